// CVNeuralNetwork_60610578482012
// MI455X (gfx1250) — compile-verified
//
#include <hip/hip_runtime.h>
#include <math.h>

// ---------------------------------------------------------------------------
// CV quantum neural network forward: N_MODES=6, CUTOFF=3, BATCH=1024, LAYERS=6
// One wave32 per batch element; 729-amplitude complex state ping-pongs between
// two LDS buffers. All expm gates are REAL (generators real), so each gate is
// two real matmuls (re/im) -> V_WMMA_F32_16X16X4_F32 with compile-time strides.
// Guards are branch-free: invalid reads hit a zero-pad LDS word, invalid
// writes hit a dump word (address cndmask instead of exec branching).
// ---------------------------------------------------------------------------

#define NMODES   6
#define BATCH_N  1024
#define NLAYERS  6
#define DIM      729            // 3^6
#define NIF      35             // params per interferometer layer
#define PLANE    736            // 729 amplitudes + pad words
#define ZERO_IDX 729            // always-zero word (safe read target)
#define DUMP_IDX 730            // write sink for invalid rows

typedef float v2f __attribute__((ext_vector_type(2)));
typedef float v8f __attribute__((ext_vector_type(8)));

// workspace layout (float offsets)
#define OFF_BS     0                        // 180 gates * 81 floats
#define OFF_SQ     (180 * 81)               // 36 gates * 9
#define OFF_DSP    (OFF_SQ + 36 * 9)        // 36 gates * 9
#define OFF_DINIT  (OFF_DSP + 36 * 9)       // 6144 gates * 9
#define WS_FLOATS  (OFF_DINIT + 6144 * 9)

// ---------------------------------------------------------------------------
// Gate preparation: expm of small real matrices (scaling-and-squaring Taylor)
// ---------------------------------------------------------------------------
template <int N>
__device__ __forceinline__ void mm_small(const float* A, const float* B, float* C) {
#pragma unroll
    for (int i = 0; i < N; ++i)
#pragma unroll
        for (int j = 0; j < N; ++j) {
            float s = 0.f;
#pragma unroll
            for (int k = 0; k < N; ++k) s += A[i * N + k] * B[k * N + j];
            C[i * N + j] = s;
        }
}

template <int N>
__device__ void expm_gen(const float* G, float theta, float* __restrict__ R) {
    float M[N * N], T[N * N], P[N * N], Q[N * N];
    const float s = theta * 0.015625f;            // theta / 64
#pragma unroll
    for (int i = 0; i < N * N; ++i) M[i] = G[i] * s;
#pragma unroll
    for (int i = 0; i < N * N; ++i) {
        float id = ((i % (N + 1)) == 0) ? 1.f : 0.f;
        T[i] = id + M[i];
        P[i] = M[i];
    }
    for (int t = 2; t <= 9; ++t) {                // Taylor terms
        mm_small<N>(P, M, Q);
        float inv = 1.f / (float)t;
#pragma unroll
        for (int i = 0; i < N * N; ++i) { P[i] = Q[i] * inv; T[i] += P[i]; }
    }
    for (int sq = 0; sq < 6; ++sq) {              // (e^{G/64})^64
        mm_small<N>(T, T, Q);
#pragma unroll
        for (int i = 0; i < N * N; ++i) T[i] = Q[i];
    }
#pragma unroll
    for (int i = 0; i < N * N; ++i) R[i] = T[i];
}

__device__ __forceinline__ void build_a(float* Am, float* Ad) {
    const float r2 = 1.41421356237309504880f;
#pragma unroll
    for (int i = 0; i < 9; ++i) { Am[i] = 0.f; Ad[i] = 0.f; }
    Am[0 * 3 + 1] = 1.f; Am[1 * 3 + 2] = r2;      // annihilation
    Ad[1 * 3 + 0] = 1.f; Ad[2 * 3 + 1] = r2;      // creation
}

__global__ __launch_bounds__(64) void cv_prep_kernel(const float* __restrict__ x,
                                                     const float* __restrict__ th1,
                                                     const float* __restrict__ th2,
                                                     const float* __restrict__ sqr,
                                                     const float* __restrict__ dspr,
                                                     float* __restrict__ ws) {
    int tid = blockIdx.x * blockDim.x + threadIdx.x;
    if (tid >= 180 + 36 + 36 + BATCH_N * NMODES) return;
    float Am[9], Ad[9];
    build_a(Am, Ad);

    if (tid < 180) {                               // beamsplitter 9x9 gates
        float BS[81];
#pragma unroll
        for (int i = 0; i < 3; ++i)
#pragma unroll
            for (int j = 0; j < 3; ++j)
#pragma unroll
                for (int k = 0; k < 3; ++k)
#pragma unroll
                    for (int l = 0; l < 3; ++l)
                        BS[(3 * i + k) * 9 + (3 * j + l)] =
                            Ad[i * 3 + j] * Am[k * 3 + l] - Am[i * 3 + j] * Ad[k * 3 + l];
        int L = tid / 30, rem = tid % 30, half = rem / 15, n = rem % 15;
        const float* th = half ? th2 : th1;
        expm_gen<9>(BS, th[L * NIF + n], ws + OFF_BS + tid * 81);
    } else if (tid < 216) {                        // squeezing: expm(0.25 r (AA - AdAd))
        float AA[9], DD[9], SQ[9];
        mm_small<3>(Am, Am, AA);
        mm_small<3>(Ad, Ad, DD);
#pragma unroll
        for (int i = 0; i < 9; ++i) SQ[i] = AA[i] - DD[i];
        int g = tid - 180;
        expm_gen<3>(SQ, 0.25f * sqr[g], ws + OFF_SQ + g * 9);
    } else {                                       // displacement: expm(r (Ad - A))
        float DG[9];
#pragma unroll
        for (int i = 0; i < 9; ++i) DG[i] = Ad[i] - Am[i];
        if (tid < 252) {
            int g = tid - 216;
            expm_gen<3>(DG, dspr[g], ws + OFF_DSP + g * 9);
        } else {
            int g = tid - 252;                     // g = b*6 + m, matches x layout
            expm_gen<3>(DG, x[g], ws + OFF_DINIT + g * 9);
        }
    }
}

// ---------------------------------------------------------------------------
// Evolution kernel building blocks (all strides compile-time)
// ---------------------------------------------------------------------------

// 3x3 real gate, amplitude stride S
template <int S>
__device__ __forceinline__ void apply1(const float* __restrict__ U9,
                                       const float* sre, const float* sim,
                                       float* dre, float* dim, int lane) {
    float u[9];
#pragma unroll
    for (int k = 0; k < 9; ++k) u[k] = U9[k];
    for (int idx = lane; idx < DIM; idx += 32) {
        int i = (idx / S) % 3;
        int base = idx - i * S;
        float re = 0.f, im = 0.f;
#pragma unroll
        for (int j = 0; j < 3; ++j) {
            float w = (i == 0) ? u[j] : ((i == 1) ? u[3 + j] : u[6 + j]);
            re += w * sre[base + j * S];
            im += w * sim[base + j * S];
        }
        dre[idx] = re;
        dim[idx] = im;
    }
}

// 9x9 real beamsplitter on adjacent modes; q-stride S3 (compile-time).
// Flat index = (o/S3)*(9*S3) + q*S3 + (o%S3), o in [0,81), q in [0,9).
// 5 full 16-wide WMMA tiles cover o in [0,80); o=80 handled scalar.
template <int S3>
__device__ __forceinline__ void applyBS(const float* __restrict__ U81,
                                        const float* sre, const float* sim,
                                        float* dre, float* dim, int lane) {
    const int half = lane >> 4;            // K rows {0,1} vs {2,3} of each chunk
    const int lrow = lane & 15;
    const int kb = half * 2;
    const bool rv = (lrow < 9);
    const int rbase = (rv ? lrow : 0) * 9; // clamp -> unconditional loads
    float a0 = U81[rbase + kb + 0];
    float a1 = U81[rbase + kb + 1];
    float a4 = U81[rbase + kb + 4];
    float a5 = U81[rbase + kb + 5];
    float a8 = U81[rbase + 8];
    v2f A0, A1, A2;
    A0.x = rv ? a0 : 0.f;
    A0.y = rv ? a1 : 0.f;
    A1.x = rv ? a4 : 0.f;
    A1.y = rv ? a5 : 0.f;
    A2.x = (rv && half == 0) ? a8 : 0.f;   // q = kb+8 valid only for half==0
    A2.y = 0.f;

#pragma unroll
    for (int tile = 0; tile < 5; ++tile) { // o = tile*16 + lrow in [0,80): always valid
        int o = tile * 16 + lrow;
        int obase = (o / S3) * (9 * S3) + (o % S3);    // const-div -> shifts/mads
        int aq8 = (half == 0) ? (obase + 8 * S3) : ZERO_IDX;   // cndmask, no branch

        v2f Br0, Br1, Br2, Bi0, Bi1, Bi2;
        Br0.x = sre[obase + (kb + 0) * S3];
        Br0.y = sre[obase + (kb + 1) * S3];
        Br1.x = sre[obase + (kb + 4) * S3];
        Br1.y = sre[obase + (kb + 5) * S3];
        Br2.x = sre[aq8];
        Br2.y = 0.f;
        Bi0.x = sim[obase + (kb + 0) * S3];
        Bi0.y = sim[obase + (kb + 1) * S3];
        Bi1.x = sim[obase + (kb + 4) * S3];
        Bi1.y = sim[obase + (kb + 5) * S3];
        Bi2.x = sim[aq8];
        Bi2.y = 0.f;

        v8f Cre = {0.f, 0.f, 0.f, 0.f, 0.f, 0.f, 0.f, 0.f};
        v8f Cim = {0.f, 0.f, 0.f, 0.f, 0.f, 0.f, 0.f, 0.f};
        Cre = __builtin_amdgcn_wmma_f32_16x16x4_f32(false, A0, false, Br0, (short)0, Cre, false, false);
        Cre = __builtin_amdgcn_wmma_f32_16x16x4_f32(false, A1, false, Br1, (short)0, Cre, false, false);
        Cre = __builtin_amdgcn_wmma_f32_16x16x4_f32(false, A2, false, Br2, (short)0, Cre, false, false);
        Cim = __builtin_amdgcn_wmma_f32_16x16x4_f32(false, A0, false, Bi0, (short)0, Cim, false, false);
        Cim = __builtin_amdgcn_wmma_f32_16x16x4_f32(false, A1, false, Bi1, (short)0, Cim, false, false);
        Cim = __builtin_amdgcn_wmma_f32_16x16x4_f32(false, A2, false, Bi2, (short)0, Cim, false, false);

        // D layout: VGPR v -> (M = v + 8*half, N = lrow). Invalid rows -> dump word.
#pragma unroll
        for (int v = 0; v < 8; ++v) {
            int p = v + half * 8;
            int addr = (v == 0 || half == 0) ? (obase + p * S3) : DUMP_IDX;
            dre[addr] = Cre[v];
            dim[addr] = Cim[v];
        }
    }

    // o = 80 column, scalar on lanes 0..8 (outside WMMA region: EXEC full there)
    {
        constexpr int ob80 = (80 / S3) * (9 * S3) + (80 % S3);
        if (lane < 9) {
            float re = 0.f, im = 0.f;
#pragma unroll
            for (int q = 0; q < 9; ++q) {
                float w = U81[lane * 9 + q];
                re += w * sre[ob80 + q * S3];
                im += w * sim[ob80 + q * S3];
            }
            dre[ob80 + lane * S3] = re;
            dim[ob80 + lane * S3] = im;
        }
    }
}

template <int S>
__device__ __forceinline__ void applyPhaseLin(float phi, float* re, float* im, int lane) {
    for (int idx = lane; idx < DIM; idx += 32) {
        int d = (idx / S) % 3;
        float ang = phi * (float)d;
        float s, c;
        __sincosf(ang, &s, &c);
        float r0 = re[idx], i0 = im[idx];
        re[idx] = r0 * c - i0 * s;
        im[idx] = r0 * s + i0 * c;
    }
}

template <int S>
__device__ __forceinline__ void applyKerr(float coef, float* re, float* im, int lane) {
    for (int idx = lane; idx < DIM; idx += 32) {
        int d = (idx / S) % 3;
        float ang = coef * (float)(d * d);
        float s, c;
        __sincosf(ang, &s, &c);
        float r0 = re[idx], i0 = im[idx];
        re[idx] = r0 * c - i0 * s;
        im[idx] = r0 * s + i0 * c;
    }
}

template <int S>
__device__ __forceinline__ float expectX(const float* re, const float* im, int lane) {
    const float r2 = 1.41421356237309504880f;
    float acc = 0.f;
    for (int idx = lane; idx < DIM; idx += 32) {
        int i = (idx / S) % 3;
        int base = idx - i * S;
        float xr, xi;
        if (i == 0)      { xr = re[base + S];                     xi = im[base + S]; }
        else if (i == 1) { xr = re[base] + r2 * re[base + 2 * S]; xi = im[base] + r2 * im[base + 2 * S]; }
        else             { xr = r2 * re[base + S];                xi = r2 * im[base + S]; }
        acc += re[idx] * xr + im[idx] * xi;
    }
#pragma unroll
    for (int off = 16; off > 0; off >>= 1) acc += __shfl_down(acc, off, 32);
    return acc;   // valid on lane 0
}

// gate + ping-pong swap + barrier (single-wave WG -> barrier is S_NOP in HW)
#define GATE1(PTR, S) do { apply1<S>((PTR), sre, sim, dre, dim, lane);            \
        { float* t = sre; sre = dre; dre = t; t = sim; sim = dim; dim = t; }      \
        __syncthreads(); } while (0)
#define GATEBS(PTR, S3) do { applyBS<S3>((PTR), sre, sim, dre, dim, lane);        \
        { float* t = sre; sre = dre; dre = t; t = sim; sim = dim; dim = t; }      \
        __syncthreads(); } while (0)

__global__ __launch_bounds__(32) void cv_evolve_kernel(const float* __restrict__ th1,
                                                       const float* __restrict__ th2,
                                                       const float* __restrict__ kerr,
                                                       const float* __restrict__ ws,
                                                       float* __restrict__ out) {
    __shared__ float Sh[2][2][PLANE];    // [buffer][re/im][amplitude+pads] = 11.8 KB
    const int lane = threadIdx.x;
    const int b = blockIdx.x;
    float* sre = Sh[0][0];
    float* sim = Sh[0][1];
    float* dre = Sh[1][0];
    float* dim = Sh[1][1];

    for (int idx = lane; idx < PLANE; idx += 32) {
        Sh[0][0][idx] = 0.f; Sh[0][1][idx] = 0.f;
        Sh[1][0][idx] = 0.f; Sh[1][1][idx] = 0.f;
    }
    __syncthreads();
    if (lane == 0) sre[0] = 1.f;         // |00...0>
    __syncthreads();

    // initial batch-dependent displacement gates (modes 0..5, strides 243..1)
    {
        const float* di = ws + OFF_DINIT + b * 54;
        GATE1(di + 0, 243); GATE1(di + 9, 81); GATE1(di + 18, 27);
        GATE1(di + 27, 9);  GATE1(di + 36, 3); GATE1(di + 45, 1);
    }

#pragma unroll 1
    for (int L = 0; L < NLAYERS; ++L) {
        // ---- interferometer 1: 15 beamsplitters (k = 0,2,4,1,3 x3) + 5 phases
        {
            const float* g = ws + OFF_BS + (L * 30) * 81;
#pragma unroll 1
            for (int rep = 0; rep < 3; ++rep, g += 5 * 81) {
                GATEBS(g + 0 * 81, 81);   // k=0
                GATEBS(g + 1 * 81, 9);    // k=2
                GATEBS(g + 2 * 81, 1);    // k=4
                GATEBS(g + 3 * 81, 27);   // k=1
                GATEBS(g + 4 * 81, 3);    // k=3
            }
            const float* rp = th1 + L * NIF + 29;
            applyPhaseLin<243>(rp[0], sre, sim, lane); __syncthreads();
            applyPhaseLin<81>(rp[1], sre, sim, lane);  __syncthreads();
            applyPhaseLin<27>(rp[2], sre, sim, lane);  __syncthreads();
            applyPhaseLin<9>(rp[3], sre, sim, lane);   __syncthreads();
            applyPhaseLin<3>(rp[4], sre, sim, lane);   __syncthreads();
        }
        // ---- squeezing
        {
            const float* g = ws + OFF_SQ + L * 54;
            GATE1(g + 0, 243); GATE1(g + 9, 81); GATE1(g + 18, 27);
            GATE1(g + 27, 9);  GATE1(g + 36, 3); GATE1(g + 45, 1);
        }
        // ---- interferometer 2
        {
            const float* g = ws + OFF_BS + (L * 30 + 15) * 81;
#pragma unroll 1
            for (int rep = 0; rep < 3; ++rep, g += 5 * 81) {
                GATEBS(g + 0 * 81, 81);
                GATEBS(g + 1 * 81, 9);
                GATEBS(g + 2 * 81, 1);
                GATEBS(g + 3 * 81, 27);
                GATEBS(g + 4 * 81, 3);
            }
            const float* rp = th2 + L * NIF + 29;
            applyPhaseLin<243>(rp[0], sre, sim, lane); __syncthreads();
            applyPhaseLin<81>(rp[1], sre, sim, lane);  __syncthreads();
            applyPhaseLin<27>(rp[2], sre, sim, lane);  __syncthreads();
            applyPhaseLin<9>(rp[3], sre, sim, lane);   __syncthreads();
            applyPhaseLin<3>(rp[4], sre, sim, lane);   __syncthreads();
        }
        // ---- displacement + Kerr (interleaved per mode, as in reference)
        {
            const float* g = ws + OFF_DSP + L * 54;
            const float* kk = kerr + L * NMODES;
            GATE1(g + 0, 243);  applyKerr<243>(kk[0] * 0.001f, sre, sim, lane); __syncthreads();
            GATE1(g + 9, 81);   applyKerr<81>(kk[1] * 0.001f, sre, sim, lane);  __syncthreads();
            GATE1(g + 18, 27);  applyKerr<27>(kk[2] * 0.001f, sre, sim, lane);  __syncthreads();
            GATE1(g + 27, 9);   applyKerr<9>(kk[3] * 0.001f, sre, sim, lane);   __syncthreads();
            GATE1(g + 36, 3);   applyKerr<3>(kk[4] * 0.001f, sre, sim, lane);   __syncthreads();
            GATE1(g + 45, 1);   applyKerr<1>(kk[5] * 0.001f, sre, sim, lane);   __syncthreads();
        }
    }

    // ---- expectation values <X_m>
    {
        float e;
        e = expectX<243>(sre, sim, lane); if (lane == 0) out[b * NMODES + 0] = e;
        e = expectX<81>(sre, sim, lane);  if (lane == 0) out[b * NMODES + 1] = e;
        e = expectX<27>(sre, sim, lane);  if (lane == 0) out[b * NMODES + 2] = e;
        e = expectX<9>(sre, sim, lane);   if (lane == 0) out[b * NMODES + 3] = e;
        e = expectX<3>(sre, sim, lane);   if (lane == 0) out[b * NMODES + 4] = e;
        e = expectX<1>(sre, sim, lane);   if (lane == 0) out[b * NMODES + 5] = e;
    }
}

// ---------------------------------------------------------------------------
extern "C" void kernel_launch(void* const* d_in, const int* in_sizes, int n_in,
                              void* d_out, int out_size, void* d_ws, size_t ws_size,
                              hipStream_t stream) {
    const float* x    = (const float*)d_in[0];
    const float* th1  = (const float*)d_in[1];
    const float* th2  = (const float*)d_in[2];
    const float* sqr  = (const float*)d_in[3];
    const float* dspr = (const float*)d_in[4];
    const float* kerr = (const float*)d_in[5];
    float* ws  = (float*)d_ws;
    float* out = (float*)d_out;
    if (ws_size < (size_t)WS_FLOATS * sizeof(float)) return;

    const int n_gates = 180 + 36 + 36 + BATCH_N * NMODES;   // 6396
    cv_prep_kernel<<<(n_gates + 63) / 64, 64, 0, stream>>>(x, th1, th2, sqr, dspr, ws);
    cv_evolve_kernel<<<BATCH_N, 32, 0, stream>>>(th1, th2, kerr, ws, out);
}